// LowPassFilter1D_62440234549674
// MI455X (gfx1250) — compile-verified
//
#include <hip/hip_runtime.h>

typedef __attribute__((ext_vector_type(2))) float v2f;
typedef __attribute__((ext_vector_type(8))) float v8f;

#define KSIZE   127
#define PAD     16          // wp[j+PAD] = w[j]; zero elsewhere
#define WP_SIZE 176         // wp indices used: [0, 158]
#define NCHUNK  36          // t in [-64, 79], chunks of 4

// Each wave computes 256-output tiles: D(16x16) = sum over 36 chunks of
// A(16x4: signal) x B(4x16: Toeplitz filter slice), V_WMMA_F32_16X16X4_F32.
__global__ __launch_bounds__(256) void fir_wmma_kernel(
    const float* __restrict__ x, const float* __restrict__ w,
    float* __restrict__ out, int n, int numTiles, int totalWaves)
{
    __shared__ float wp[WP_SIZE];
    for (int t = threadIdx.x; t < WP_SIZE; t += blockDim.x) {
        int j = t - PAD;
        wp[t] = (j >= 0 && j < KSIZE) ? w[j] : 0.0f;
    }
    __syncthreads();

    const int lane = threadIdx.x & 31;
    const int wave = blockIdx.x * (blockDim.x >> 5) + (threadIdx.x >> 5);

    const int m  = lane & 15;          // A row (output sub-row)
    const int nn = lane & 15;          // B column (output column)
    const int k0 = (lane >> 4) << 1;   // K-pair held by this half-wave

    // 36 loop-invariant B fragments (filter Toeplitz slices), kept in VGPRs
    // for the whole grid-stride loop.  B[k][n] = wp[t0 + k + 63 - n + PAD].
    v2f bfrag[NCHUNK];
#pragma unroll
    for (int c = 0; c < NCHUNK; ++c) {
        int i0 = (-64 + 4 * c) + k0 + 79 - nn;   // in [0, 158]
        v2f b;
        b.x = wp[i0];
        b.y = wp[i0 + 1];
        bfrag[c] = b;
    }

    for (int tile = wave; tile < numTiles; tile += totalWaves) {
        long base = (long)tile * 256;
        // interior: every x index in [base-64, base+320) is in-bounds
        if (base >= 64 && base + 320 <= (long)n) {
            // A[m][k] = x[base + 16*m + t0 + k]; per lane: float2 at k0, 8B aligned
            const float* xa = x + base + 16 * m + k0 - 64;
            v8f acc = {};
#pragma unroll
            for (int c = 0; c < NCHUNK; ++c) {
                v2f a = *(const v2f*)(xa + 4 * c);
                acc = __builtin_amdgcn_wmma_f32_16x16x4_f32(
                    false, a, false, bfrag[c], (short)0, acc, false, false);
            }
            // D layout: VGPR r -> lanes 0-15: out[base+16r+lane] (M=r),
            //                     lanes 16-31: out[base+16r+lane+112] (M=8+r)
            long sb = base + lane + (lane >= 16 ? 112 : 0);
#pragma unroll
            for (int r = 0; r < 8; ++r)
                out[sb + 16 * r] = acc[r];
        } else {
            // boundary tiles: scalar path with zero padding (wave-uniform branch)
            for (int q = 0; q < 8; ++q) {
                long p = base + q * 32 + lane;
                if (p >= (long)n) continue;
                float s = 0.0f;
                for (int j = 0; j < KSIZE; ++j) {
                    long idx = p + j - 63;
                    if (idx >= 0 && idx < (long)n) s += x[idx] * wp[j + PAD];
                }
                out[p] = s;
            }
        }
    }
}

extern "C" void kernel_launch(void* const* d_in, const int* in_sizes, int n_in,
                              void* d_out, int out_size, void* d_ws, size_t ws_size,
                              hipStream_t stream) {
    const float* x = (const float*)d_in[0];
    const float* w = (const float*)d_in[1];
    float* out = (float*)d_out;
    int n = in_sizes[0];

    int numTiles = (out_size + 255) / 256;
    const int threads = 256;                 // 8 waves/block (wave32)
    const int wavesPerBlock = threads / 32;
    int blocks = 2048;                       // grid-stride: ~8 tiles/wave at N=2^25
    if ((long)blocks * wavesPerBlock > (long)numTiles) {
        blocks = (numTiles + wavesPerBlock - 1) / wavesPerBlock;
        if (blocks < 1) blocks = 1;
    }
    int totalWaves = blocks * wavesPerBlock;

    fir_wmma_kernel<<<blocks, threads, 0, stream>>>(x, w, out, n, numTiles, totalWaves);
}